// NGCF_52527450030532
// MI455X (gfx1250) — compile-verified
//
#include <hip/hip_runtime.h>
#include <math.h>

#define N_USERS 50000
#define N_ITEMS 50000
#define N_NODES 100000
#define EMB 64
#define NLAYERS 3
#define NE 1600000
#define BATCH 16384
#define REG_L2 1e-5f
#define SLOPE 0.01f
#define EPS_N 1e-12f

#define RS_B 160  // floats per k-pair row of B in LDS (640B: halves hit disjoint bank sets)
#define RS_A 132  // floats per A row in LDS (bank-conflict-free strided v2f reads)
#define B_FLOATS (64 * RS_B)          // 10240
#define A_FLOATS (8 * 16 * RS_A)      // 16896
#define SMEM_FLOATS (B_FLOATS + A_FLOATS + 64)
#define SMEM_BYTES (SMEM_FLOATS * 4)  // 108,800 B of the 320KB WGP LDS

typedef float v2f __attribute__((ext_vector_type(2)));
typedef float v8f __attribute__((ext_vector_type(8)));

// ---------------------------------------------------------------------------
// K2: SpMM scatter: ws[row] += val * ego[col].  16 lanes per edge, float4 each.
// ws (25.6MB) is L2-resident (192MB), so f32 atomics resolve at L2 rate.
// ---------------------------------------------------------------------------
__global__ void __launch_bounds__(256)
spmm_scatter(const int* __restrict__ rows, const int* __restrict__ cols,
             const float* __restrict__ vals, const float* __restrict__ ego,
             float* __restrict__ ws) {
  long long gid = (long long)blockIdx.x * blockDim.x + threadIdx.x;
  long long e = gid >> 4;
  int q = (int)(gid & 15);
  if (e >= NE) return;
  int c = cols[e];
  int r = rows[e];
  float v = vals[e];
  float4 m = ((const float4*)(ego + (long long)c * EMB))[q];
  float* dst = ws + (long long)r * EMB + q * 4;
  atomicAdd(dst + 0, v * m.x);
  atomicAdd(dst + 1, v * m.y);
  atomicAdd(dst + 2, v * m.z);
  atomicAdd(dst + 3, v * m.w);
}

// ---------------------------------------------------------------------------
// K3: fused dense layer. One wave per 16-row tile:
//   A(16x128) = [ws_row | ws_row*ego_row] staged in LDS via coalesced b128,
//   B(128x64) = [W1^T ; W2^T] pair-interleaved in LDS (one ds_load_b64 / WMMA),
//   T = A@B + b via V_WMMA_F32_16X16X4_F32 (4 N-tiles x 32 K-steps),
//   then LeakyReLU + per-row L2 normalize (half-wave shfl reduction).
// WMMA f32 16x16x4 layouts (ISA 7.12.2):
//   A: lane L: m=L&15, kh=L>>4; a = {A[m][kb+2kh], A[m][kb+2kh+1]}
//   B: lane L: n=L&15;          b = {B[kb+2kh][n], B[kb+2kh+1][n]}
//   C: VGPR r: lanes0-15 -> M=r, lanes16-31 -> M=r+8; N = (L&15) + 16*nt
// ---------------------------------------------------------------------------
__global__ void __launch_bounds__(256)
layer_wmma(const float* __restrict__ ego, const float* __restrict__ ws,
           const float* __restrict__ W1w, const float* __restrict__ W1b,
           const float* __restrict__ W2w, const float* __restrict__ W2b,
           float* __restrict__ out) {
  extern __shared__ float smem[];
  float* Bsm = smem;                 // [64 kpairs][RS_B]: pair-interleaved B
  float* Asm = smem + B_FLOATS;      // [8 waves][16 rows][RS_A]
  float* bias = Asm + A_FLOATS;      // [64]

  // Stage B pair-interleaved: Bp[p][2n+(kk&1)] = B[kk][n],
  // where B[kk][n] = (kk<64) ? W1[n][kk] : W2[n][kk-64]   (t = ws@W1^T + aff@W2^T)
  for (int idx = threadIdx.x; idx < 128 * 64; idx += 256) {
    int kk = idx >> 6, n = idx & 63;
    float w = (kk < 64) ? W1w[n * 64 + kk] : W2w[n * 64 + (kk - 64)];
    Bsm[(kk >> 1) * RS_B + n * 2 + (kk & 1)] = w;
  }
  if (threadIdx.x < 64) bias[threadIdx.x] = W1b[threadIdx.x] + W2b[threadIdx.x];

  int wave = threadIdx.x >> 5;
  int lane = threadIdx.x & 31;
  int tile = blockIdx.x * 8 + wave;
  bool active = (tile < N_NODES / 16);
  float* Aw = Asm + wave * 16 * RS_A;

  if (active) {
    long long base = (long long)tile * 16 * EMB;
    // Coalesced tile stage: 8 iters x (32 lanes x float4) = 2 rows per iter.
#pragma unroll
    for (int it = 0; it < 8; ++it) {
      int f = it * 32 + lane;          // float4 index within 16x64 tile
      int m = f >> 4, c4 = (f & 15) * 4;
      float4 w4 = *(const float4*)(ws + base + m * EMB + c4);
      float4 e4 = *(const float4*)(ego + base + m * EMB + c4);
      float* arow = Aw + m * RS_A;
      *(float4*)(arow + c4) = w4;                                  // ws part
      float4 a4;                                                   // aff part
      a4.x = w4.x * e4.x; a4.y = w4.y * e4.y;
      a4.z = w4.z * e4.z; a4.w = w4.w * e4.w;
      *(float4*)(arow + 64 + c4) = a4;
    }
  }
  __syncthreads();
  if (!active) return;

  int m = lane & 15;
  int kh = lane >> 4;
  int n0 = lane & 15;
  const float* arow = Aw + m * RS_A;

  v8f c0 = {}, c1 = {}, c2 = {}, c3 = {};
#pragma unroll
  for (int ks = 0; ks < 32; ++ks) {
    int kk = 4 * ks + 2 * kh;
    v2f a = *(const v2f*)(arow + kk);                 // conflict-free ds b64
    const float* bp = Bsm + (2 * ks + kh) * RS_B + n0 * 2;
    v2f b0 = *(const v2f*)(bp);
    v2f b1 = *(const v2f*)(bp + 32);
    v2f b2 = *(const v2f*)(bp + 64);
    v2f b3 = *(const v2f*)(bp + 96);
    c0 = __builtin_amdgcn_wmma_f32_16x16x4_f32(false, a, false, b0, (short)0, c0, false, false);
    c1 = __builtin_amdgcn_wmma_f32_16x16x4_f32(false, a, false, b1, (short)0, c1, false, false);
    c2 = __builtin_amdgcn_wmma_f32_16x16x4_f32(false, a, false, b2, (short)0, c2, false, false);
    c3 = __builtin_amdgcn_wmma_f32_16x16x4_f32(false, a, false, b3, (short)0, c3, false, false);
  }

  // Epilogue: bias + LeakyReLU + row L2-normalize + store.
#pragma unroll
  for (int r = 0; r < 8; ++r) {
    float t0 = c0[r] + bias[n0];
    float t1 = c1[r] + bias[n0 + 16];
    float t2 = c2[r] + bias[n0 + 32];
    float t3 = c3[r] + bias[n0 + 48];
    t0 = (t0 >= 0.f) ? t0 : SLOPE * t0;
    t1 = (t1 >= 0.f) ? t1 : SLOPE * t1;
    t2 = (t2 >= 0.f) ? t2 : SLOPE * t2;
    t3 = (t3 >= 0.f) ? t3 : SLOPE * t3;
    float s = t0 * t0 + t1 * t1 + t2 * t2 + t3 * t3;
    // reduce across the 16 lanes of this half-wave (rows r / r+8 independent)
    s += __shfl_xor(s, 1, 32);
    s += __shfl_xor(s, 2, 32);
    s += __shfl_xor(s, 4, 32);
    s += __shfl_xor(s, 8, 32);
    float inv = 1.0f / fmaxf(sqrtf(s), EPS_N);
    int grow = tile * 16 + 8 * kh + r;
    float* dst = out + (long long)grow * EMB + n0;
    dst[0]  = t0 * inv;
    dst[16] = t1 * inv;
    dst[32] = t2 * inv;
    dst[48] = t3 * inv;
  }
}

// ---------------------------------------------------------------------------
// K4: per-batch partial dots for one 64-wide pred block. Wave per triple.
// acc_ui += <u,i>, acc_uj += <u,j>, acc_l2 += |u|^2+|i|^2+|j|^2
// ---------------------------------------------------------------------------
__global__ void __launch_bounds__(256)
batch_acc(const int* __restrict__ u, const int* __restrict__ ii,
          const int* __restrict__ jj, const float* __restrict__ blk,
          float* __restrict__ acc_ui, float* __restrict__ acc_uj,
          float* __restrict__ acc_l2) {
  int t = blockIdx.x * blockDim.x + threadIdx.x;
  int b = t >> 5, lane = t & 31;
  if (b >= BATCH) return;
  long long ru = (long long)u[b] * EMB;
  long long rp = (long long)(N_USERS + ii[b]) * EMB;
  long long rn = (long long)(N_USERS + jj[b]) * EMB;
  float2 ue = *(const float2*)(blk + ru + lane * 2);
  float2 pe = *(const float2*)(blk + rp + lane * 2);
  float2 ne = *(const float2*)(blk + rn + lane * 2);
  float dui = ue.x * pe.x + ue.y * pe.y;
  float duj = ue.x * ne.x + ue.y * ne.y;
  float l2 = ue.x * ue.x + ue.y * ue.y + pe.x * pe.x + pe.y * pe.y +
             ne.x * ne.x + ne.y * ne.y;
#pragma unroll
  for (int off = 16; off; off >>= 1) {
    dui += __shfl_xor(dui, off, 32);
    duj += __shfl_xor(duj, off, 32);
    l2 += __shfl_xor(l2, off, 32);
  }
  if (lane == 0) {
    acc_ui[b] += dui;
    acc_uj[b] += duj;
    acc_l2[b] += l2;
  }
}

// ---------------------------------------------------------------------------
// K5: final scalar: -mean(logsigmoid(ui-uj)) + REG * sum(l2)/(2*BATCH)
// ---------------------------------------------------------------------------
__global__ void __launch_bounds__(256)
final_loss(const float* __restrict__ acc_ui, const float* __restrict__ acc_uj,
           const float* __restrict__ acc_l2, float* __restrict__ out) {
  __shared__ float s1[256], s2[256];
  float a = 0.f, l = 0.f;
  for (int i = threadIdx.x; i < BATCH; i += 256) {
    float x = acc_ui[i] - acc_uj[i];
    float ls = (x >= 0.f) ? -log1pf(expf(-x)) : (x - log1pf(expf(x)));
    a += ls;
    l += acc_l2[i];
  }
  s1[threadIdx.x] = a;
  s2[threadIdx.x] = l;
  __syncthreads();
  for (int o = 128; o; o >>= 1) {
    if (threadIdx.x < o) {
      s1[threadIdx.x] += s1[threadIdx.x + o];
      s2[threadIdx.x] += s2[threadIdx.x + o];
    }
    __syncthreads();
  }
  if (threadIdx.x == 0)
    out[0] = -(s1[0] / (float)BATCH) + REG_L2 * (s2[0] * 0.5f / (float)BATCH);
}

// ---------------------------------------------------------------------------
extern "C" void kernel_launch(void* const* d_in, const int* in_sizes, int n_in,
                              void* d_out, int out_size, void* d_ws,
                              size_t ws_size, hipStream_t stream) {
  (void)in_sizes; (void)n_in; (void)out_size; (void)ws_size;
  const int* u = (const int*)d_in[0];
  const int* ii = (const int*)d_in[1];
  const int* jj = (const int*)d_in[2];
  const int* rows = (const int*)d_in[3];
  const int* cols = (const int*)d_in[4];
  const float* vals = (const float*)d_in[5];
  const float* u_emb = (const float*)d_in[6];
  const float* i_emb = (const float*)d_in[7];
  const float* W1w = (const float*)d_in[8];
  const float* W1b = (const float*)d_in[9];
  const float* W2w = (const float*)d_in[10];
  const float* W2b = (const float*)d_in[11];

  const size_t NODEF = (size_t)N_NODES * EMB;
  float* egoA = (float*)d_ws;
  float* egoB = egoA + NODEF;
  float* wsb = egoB + NODEF;
  float* aui = wsb + NODEF;
  float* auj = aui + BATCH;
  float* al2 = auj + BATCH;

  // zero batch accumulators (3 contiguous arrays)
  hipMemsetAsync(aui, 0, 3 * BATCH * sizeof(float), stream);
  // ego0 = concat(u_emb, i_emb)
  hipMemcpyAsync(egoA, u_emb, (size_t)N_USERS * EMB * sizeof(float),
                 hipMemcpyDeviceToDevice, stream);
  hipMemcpyAsync(egoA + (size_t)N_USERS * EMB, i_emb,
                 (size_t)N_ITEMS * EMB * sizeof(float),
                 hipMemcpyDeviceToDevice, stream);

  dim3 accGrid((BATCH * 32 + 255) / 256);
  batch_acc<<<accGrid, 256, 0, stream>>>(u, ii, jj, egoA, aui, auj, al2);

  float* cur = egoA;
  float* nxt = egoB;
  for (int k = 0; k < NLAYERS; ++k) {
    hipMemsetAsync(wsb, 0, NODEF * sizeof(float), stream);
    long long nth = (long long)NE * 16;
    spmm_scatter<<<(unsigned)((nth + 255) / 256), 256, 0, stream>>>(
        rows, cols, vals, cur, wsb);
    int ntiles = N_NODES / 16;  // 6250
    layer_wmma<<<(ntiles + 7) / 8, 256, SMEM_BYTES, stream>>>(
        cur, wsb, W1w + (size_t)k * EMB * EMB, W1b + (size_t)k * EMB,
        W2w + (size_t)k * EMB * EMB, W2b + (size_t)k * EMB, nxt);
    batch_acc<<<accGrid, 256, 0, stream>>>(u, ii, jj, nxt, aui, auj, al2);
    float* tmp = cur; cur = nxt; nxt = tmp;
  }

  final_loss<<<1, 256, 0, stream>>>(aui, auj, al2, (float*)d_out);
}